// SplineActivation_7292854469239
// MI455X (gfx1250) — compile-verified
//
#include <hip/hip_runtime.h>

// Problem constants (from reference)
#define INPUT_DIM 4096
#define BATCH     4096
#define NB        7            // basis / coeffs per column

// Tiling
#define COLS_PER_BLOCK 1024
#define THREADS        256
#define CPT            4       // columns per thread (float4 lane)
#define GRID_Y         128
#define ROWS_PER_BLOCK (BATCH / GRID_Y)   // 32

typedef float        f4    __attribute__((ext_vector_type(4)));
typedef unsigned int u32x4 __attribute__((ext_vector_type(4)));
typedef int          i32x8 __attribute__((ext_vector_type(8)));
typedef int          i32x4 __attribute__((ext_vector_type(4)));

// Uniform cubic B-spline on grid [-1,1], h=0.5, extended knots.
// Only 4 bases are non-zero: indices j..j+3 with j = clamp(floor((x+1)/h),0,3),
// weights are the cardinal cubic B-spline weights of u = (x+1)/h - j.
__device__ __forceinline__ float spline_eval(float xv, const float* __restrict__ c) {
  float y  = __builtin_fmaf(xv, 2.0f, 2.0f);   // (x+1)/h in [0,4)
  y        = fminf(fmaxf(y, 0.0f), 4.0f);      // -> v_med3 style clamp
  float jf = fminf(floorf(y), 3.0f);
  float u  = y - jf;
  int   ji = (int)jf;                          // 0..3

  // Slide the 4-wide coefficient window c[ji..ji+3] with 9 v_cndmask (2-level).
  bool ge2 = ji >= 2;
  float a0 = ge2 ? c[2] : c[0];
  float a1 = ge2 ? c[3] : c[1];
  float a2 = ge2 ? c[4] : c[2];
  float a3 = ge2 ? c[5] : c[3];
  float a4 = ge2 ? c[6] : c[4];
  bool odd = (ji & 1) != 0;
  float b0 = odd ? a1 : a0;
  float b1 = odd ? a2 : a1;
  float b2 = odd ? a3 : a2;
  float b3 = odd ? a4 : a3;

  // Cardinal cubic weights (already divided by 6).
  float u2 = u * u;
  float u3 = u2 * u;
  float om = 1.0f - u;
  float w0 = om * om * om * (1.0f / 6.0f);
  float w3 = u3 * (1.0f / 6.0f);
  float w1 = __builtin_fmaf(0.5f, u3, __builtin_fmaf(-1.0f, u2, 2.0f / 3.0f));
  float w2 = __builtin_fmaf(-0.5f, u3,
             __builtin_fmaf(0.5f, u2, __builtin_fmaf(0.5f, u, 1.0f / 6.0f)));

  return __builtin_fmaf(b0, w0,
         __builtin_fmaf(b1, w1,
         __builtin_fmaf(b2, w2, b3 * w3)));
}

__global__ __launch_bounds__(THREADS) void spline_act_kernel(
    const float* __restrict__ x, const float* __restrict__ coeffs,
    float* __restrict__ out) {
  __shared__ float scoef[COLS_PER_BLOCK * NB];   // 28 KB coeff tile

  const int tile_d0 = (int)blockIdx.x * COLS_PER_BLOCK;

  // ---- TDM: one 1-D tensor DMA pulls this block's coeff tile into LDS ----
  // Tile = 1024 columns * 7 f32 = 7168 contiguous elements (28672 B).
  if (threadIdx.x < 32) {  // wave 0 only (EXEC all-ones within it)
    unsigned long long ga =
        reinterpret_cast<unsigned long long>(coeffs + (size_t)tile_d0 * NB);
    const unsigned NE = COLS_PER_BLOCK * NB;     // 7168 elements

    u32x4 g0;
    g0[0] = 1u;                                  // count=1, user descriptor
    g0[1] = 0u;                                  // lds_addr: scoef is the only LDS alloc -> offset 0
    g0[2] = (unsigned)(ga & 0xFFFFFFFFu);        // global_addr[31:0]
    g0[3] = (unsigned)((ga >> 32) & 0x01FFFFFFu) // global_addr[56:32]
            | (2u << 30);                        // type = 2 ("image")

    i32x8 g1;
    g1[0] = (int)(2u << 16);                     // workgroup_mask=0, data_size=2 (4 B)
    g1[1] = (int)((NE & 0xFFFFu) << 16);         // tensor_dim0[15:0]
    g1[2] = (int)((NE >> 16) | (1u << 16));      // tensor_dim0[31:16], tensor_dim1=1
    g1[3] = (int)(NE << 16);                     // tensor_dim1[31:16]=0, tile_dim0=7168
    g1[4] = 0;                                   // tile_dim1=0, tile_dim2=0 (1-D tile)
    g1[5] = (int)NE;                             // tensor_dim0_stride[31:0]
    g1[6] = 0;                                   // stride hi / dim1 stride
    g1[7] = 0;

    i32x4 gz4 = {0, 0, 0, 0};                    // groups 2/3 unused (<=2D tile)
    i32x8 gz8 = {0, 0, 0, 0, 0, 0, 0, 0};        // trailing group (clang-23 6-arg form)
    __builtin_amdgcn_tensor_load_to_lds(g0, g1, gz4, gz4, gz8, 0);
    __builtin_amdgcn_s_wait_tensorcnt(0);        // TENSORcnt == 0
  }
  __syncthreads();

  // Each thread owns 4 consecutive columns; hoist its 28 coeffs to registers.
  const int lc = (int)threadIdx.x * CPT;         // 0..1020
  float c[CPT][NB];
#pragma unroll
  for (int j = 0; j < CPT; ++j)
#pragma unroll
    for (int k = 0; k < NB; ++k)
      c[j][k] = scoef[(lc + j) * NB + k];

  const size_t dcol = (size_t)(tile_d0 + lc);
  const int row0 = (int)blockIdx.y * ROWS_PER_BLOCK;

  // Pure streaming loop: one B128 NT load + one B128 NT store per row.
#pragma unroll 4
  for (int r = row0; r < row0 + ROWS_PER_BLOCK; ++r) {
    size_t idx = (size_t)r * INPUT_DIM + dcol;
    f4* px = (f4*)(x + idx);
    // Prefetch 4 rows ahead (speculative; OOB at the tail is dropped).
    __builtin_prefetch((const char*)px + 4u * INPUT_DIM * sizeof(float), 0, 0);
    f4 xv = __builtin_nontemporal_load(px);
    f4 ov;
    ov.x = spline_eval(xv.x, c[0]);
    ov.y = spline_eval(xv.y, c[1]);
    ov.z = spline_eval(xv.z, c[2]);
    ov.w = spline_eval(xv.w, c[3]);
    __builtin_nontemporal_store(ov, (f4*)(out + idx));
  }
}

extern "C" void kernel_launch(void* const* d_in, const int* in_sizes, int n_in,
                              void* d_out, int out_size, void* d_ws, size_t ws_size,
                              hipStream_t stream) {
  (void)in_sizes; (void)n_in; (void)out_size; (void)d_ws; (void)ws_size;
  const float* x      = (const float*)d_in[0];
  const float* coeffs = (const float*)d_in[1];
  float*       out    = (float*)d_out;

  dim3 grid(INPUT_DIM / COLS_PER_BLOCK, GRID_Y);   // (4, 128) = 512 blocks
  spline_act_kernel<<<grid, THREADS, 0, stream>>>(x, coeffs, out);
}